// GCN_6279242187119
// MI455X (gfx1250) — compile-verified
//
#include <hip/hip_runtime.h>

// ---------------------------------------------------------------------------
// 3-layer GCN for MI455X (gfx1250, wave32).
//   h = X @ W                       -> V_WMMA_F32_16X16X4_F32, B via ds_load_b64
//   out = bias + h[i]*dis[i]^2      -> fused into GEMM epilogue
//         + scatter(h[src]*norm[e]) -> f32 atomics, prefetch-pipelined gather
// ---------------------------------------------------------------------------

typedef __attribute__((ext_vector_type(2))) float v2f;
typedef __attribute__((ext_vector_type(8))) float v8f;

#define N_NODES 100000
#define OFF_NORM 100096                 // N_NODES padded to 256-float multiple
#define OFF_BUFA (OFF_NORM + 1250048)   // + E padded
#define NBUF ((size_t)N_NODES * 64)

// ---------------- degree / normalization (graph fixed across layers) --------

__global__ void gcn_deg_init(float* __restrict__ deg, int n) {
  int i = blockIdx.x * blockDim.x + threadIdx.x;
  if (i < n) deg[i] = 1.0f;  // self-loop contributes 1 to in-degree
}

__global__ void gcn_deg_count(const int* __restrict__ dst, float* __restrict__ deg, int nE) {
  int e = blockIdx.x * blockDim.x + threadIdx.x;
  if (e < nE) atomicAdd(&deg[dst[e]], 1.0f);
}

__global__ void gcn_deg_rsqrt(float* __restrict__ deg, int n) {
  int i = blockIdx.x * blockDim.x + threadIdx.x;
  if (i < n) deg[i] = rsqrtf(deg[i]);  // deg >= 1 always (self-loop)
}

__global__ void gcn_edge_norm(const int* __restrict__ src, const int* __restrict__ dst,
                              const float* __restrict__ dis, float* __restrict__ nrm,
                              int nE) {
  int e = blockIdx.x * blockDim.x + threadIdx.x;
  if (e < nE) nrm[e] = dis[src[e]] * dis[dst[e]];
}

// ---------------- GEMM via WMMA f32 16x16x4, fused self-loop+bias epilogue --
// One wave owns a 16-row x NC panel: NC/16 independent WMMA accumulator
// chains (hides XDL latency); A loaded once per panel (no inter-wave
// redundancy).  W is staged TRANSPOSED in LDS (sWt[n][k], row stride
// S = K+6), so each B fragment {W[k+2h][n], W[k+2h+1][n]} is one aligned
// ds_load_b64 landing in an even VGPR pair -- no pairing v_movs.
// S == 6 (mod 64): lane bank pairs {6m+c, 6m+c+1} are all-distinct, and the
// half-1 (+2 banks) set overlaps 5 of them (6*11 == 2 mod 64): 2-way conflict.
// Epilogue writes H (for the edge gather) and OUTI = bias + H*dis^2 (the
// self-loop + bias term, fully overwriting OUTI).  OUTI may alias X: each
// wave reads/writes only its own panel, and loads precede stores.
//
// f32 WMMA operand layout (ISA 7.12.2): lane m = lane&15, half = lane>>4
//   A {X[m][k+2h], X[m][k+2h+1]};  B {W[k+2h][n], W[k+2h+1][n]}, n = m
//   D VGPR r -> row (r + 8*half), col n

template <int K, int NC, bool RELU>
__global__ __launch_bounds__(64)
void gcn_gemm_wmma(const float* X, const float* __restrict__ W,
                   float* __restrict__ H, float* OUTI,
                   const float* __restrict__ dis, const float* __restrict__ bias) {
  constexpr int S  = K + 6;              // transposed-LDS row stride (floats)
  constexpr int CT = NC / 16;            // 16-col tiles per panel
  __shared__ float sWt[NC * S];

  // Stage W (K x NC, row-major) transposed: coalesced float4 global loads,
  // 4 stride-S scalar DS stores (banks 6n+kr mod 64: conflict-free for NC=32, 2-way for NC=64).
  for (int t = threadIdx.x; t < K * NC / 4; t += 64) {
    const int kr = (t * 4) / NC;
    const int n  = (t * 4) % NC;
    const float4 w4 = *(const float4*)(W + (size_t)t * 4);
    sWt[(size_t)(n + 0) * S + kr] = w4.x;
    sWt[(size_t)(n + 1) * S + kr] = w4.y;
    sWt[(size_t)(n + 2) * S + kr] = w4.z;
    sWt[(size_t)(n + 3) * S + kr] = w4.w;
  }
  __syncthreads();

  const int lane = threadIdx.x & 31;
  const int wave = threadIdx.x >> 5;
  const int m    = lane & 15;            // A row within panel == output col n
  const int half = lane >> 4;
  const long long row0 = ((long long)blockIdx.x * 2 + wave) * 16;

  const float* xrow  = X + (size_t)(row0 + m) * K + 2 * half;
  const float* bbase = sWt + (size_t)m * S + 2 * half;

  v8f acc[CT] = {};
#pragma unroll
  for (int kk = 0; kk < K / 4; ++kk) {
    const int k = kk * 4;
    float ax = xrow[k];
    float ay = xrow[k + 1];
    if (RELU) { ax = fmaxf(ax, 0.f); ay = fmaxf(ay, 0.f); }
    const v2f a = {ax, ay};
#pragma unroll
    for (int ct = 0; ct < CT; ++ct) {
      const v2f b = *(const v2f*)(bbase + (size_t)ct * 16 * S + k);  // ds_load_b64
      acc[ct] = __builtin_amdgcn_wmma_f32_16x16x4_f32(
          false, a, false, b, (short)0, acc[ct], false, false);
    }
  }

  // Epilogue: H = acc;  OUTI = bias + acc * dis[row]^2  (self-loop + bias).
#pragma unroll
  for (int ct = 0; ct < CT; ++ct) {
    const float bv = bias[ct * 16 + m];
#pragma unroll
    for (int r = 0; r < 8; ++r) {
      const long long row = row0 + r + 8 * half;
      const float dv = dis[row];
      const float hv = acc[ct][r];
      H[(size_t)row * NC + ct * 16 + m]    = hv;
      OUTI[(size_t)row * NC + ct * 16 + m] = bv + hv * dv * dv;
    }
  }
}

// ---------------- edge scatter: out[dst] += h[src] * norm[e] -----------------
// Grid-stride, 4 features/thread (global_load_b128 gather + 4 f32 atomics),
// next iteration's gather row prefetched (global_prefetch_b8).

template <int F>
__global__ void gcn_edge_scatter(const float* __restrict__ h,
                                 const int* __restrict__ src,
                                 const int* __restrict__ dst,
                                 const float* __restrict__ nrm,
                                 float* __restrict__ out, int nE) {
  constexpr int CH = F / 4;
  const long long total  = (long long)nE * CH;
  const long long stride = (long long)gridDim.x * blockDim.x;
  for (long long idx = (long long)blockIdx.x * blockDim.x + threadIdx.x;
       idx < total; idx += stride) {
    const long long nidx = idx + stride;
    if (nidx < total) {  // prefetch next iteration's gathered row chunk
      const int ne = (int)(nidx / CH);
      __builtin_prefetch(h + (size_t)src[ne] * F + (int)(nidx % CH) * 4, 0, 0);
    }
    const int e  = (int)(idx / CH);
    const int fo = (int)(idx % CH) * 4;
    const int s  = src[e];
    const int d  = dst[e];
    const float w = nrm[e];
    const float4 v = *(const float4*)(h + (size_t)s * F + fo);
    float* o = out + (size_t)d * F + fo;
    atomicAdd(o + 0, v.x * w);
    atomicAdd(o + 1, v.y * w);
    atomicAdd(o + 2, v.z * w);
    atomicAdd(o + 3, v.w * w);
  }
}

// ---------------------------------------------------------------------------

extern "C" void kernel_launch(void* const* d_in, const int* in_sizes, int n_in,
                              void* d_out, int out_size, void* d_ws, size_t ws_size,
                              hipStream_t stream) {
  const float* x  = (const float*)d_in[0];
  const int*   ei = (const int*)d_in[1];   // [2, E]: row 0 = src, row 1 = dst
  const float* W1 = (const float*)d_in[2];
  const float* b1 = (const float*)d_in[3];
  const float* W2 = (const float*)d_in[4];
  const float* b2 = (const float*)d_in[5];
  const float* W3 = (const float*)d_in[6];
  const float* b3 = (const float*)d_in[7];
  float* out = (float*)d_out;

  const int N = N_NODES;
  const int E = in_sizes[1] / 2;
  const int* src = ei;
  const int* dst = ei + E;

  float* ws   = (float*)d_ws;
  float* dis  = ws;                    // N floats (deg -> rsqrt in place)
  float* nrm  = ws + OFF_NORM;         // E floats, reused for all 3 layers
  float* bufA = ws + OFF_BUFA;         // N*64 floats (h of current layer)
  float* bufB = bufA + NBUF;           // N*64 floats (aggregated output)

  const int TB = 256;
  const int gN = (N + TB - 1) / TB;
  const int gE = (E + TB - 1) / TB;
  const int gGemm = N / 32;            // 3125 blocks x 2 waves x 16 rows, exact
  const int gScat = 4096;              // grid-stride scatter

  // Normalization coefficients, computed once, reused by all three layers.
  gcn_deg_init <<<gN, TB, 0, stream>>>(dis, N);
  gcn_deg_count<<<gE, TB, 0, stream>>>(dst, dis, E);
  gcn_deg_rsqrt<<<gN, TB, 0, stream>>>(dis, N);
  gcn_edge_norm<<<gE, TB, 0, stream>>>(src, dst, dis, nrm, E);

  // Layer 1: GCNConv(128 -> 64).  h -> bufA, bufB = b1 + h*dis^2, then edges.
  gcn_gemm_wmma<128, 64, false><<<gGemm, 64, 0, stream>>>(x, W1, bufA, bufB, dis, b1);
  gcn_edge_scatter<64><<<gScat, TB, 0, stream>>>(bufA, src, dst, nrm, bufB, E);

  // Layer 2: GCNConv(64 -> 64), ReLU fused into A-load; OUTI overwrites bufB
  // in place (each wave touches only its own 16-row panel).
  gcn_gemm_wmma<64, 64, true><<<gGemm, 64, 0, stream>>>(bufB, W2, bufA, bufB, dis, b2);
  gcn_edge_scatter<64><<<gScat, TB, 0, stream>>>(bufA, src, dst, nrm, bufB, E);

  // Layer 3: GCNConv(64 -> 32), ReLU fused; init + aggregate straight in d_out.
  gcn_gemm_wmma<64, 32, true><<<gGemm, 64, 0, stream>>>(bufB, W3, bufA, out, dis, b3);
  gcn_edge_scatter<32><<<gScat, TB, 0, stream>>>(bufA, src, dst, nrm, out, E);
}